// Block_12695923327233
// MI455X (gfx1250) — compile-verified
//
#include <hip/hip_runtime.h>

typedef __attribute__((ext_vector_type(2))) float v2f;
typedef __attribute__((ext_vector_type(8))) float v8f;
typedef __attribute__((ext_vector_type(4))) unsigned int uint32x4;
typedef __attribute__((ext_vector_type(8))) int int32x8;
typedef __attribute__((ext_vector_type(4))) int int32x4;

#define NNODES 50000
#define NEDGES 600000
#define CH 128
#define WSTRIDE 144   // padded LDS row stride in dwords (128 data + 16 TDM pad)

// ---------------- utility: zero fp32 buffer ----------------
__global__ void zero_f32(float* __restrict__ p, int n) {
    int i = blockIdx.x * blockDim.x + threadIdx.x;
    if (i < n) p[i] = 0.0f;
}

// ---------------- scatter-add of source features onto dst rows --------------
// one wave = one edge: edge id is wave-uniform (scalarized via readfirstlane),
// 32 lanes cover the 512B feature row (float4 each) -> coalesced b128 gather,
// then 4x global_atomic_add_f32 per lane. Lane 0 optionally bumps the degree.
__global__ void scatter_kernel(const float* __restrict__ x,
                               const int* __restrict__ ei,
                               float* __restrict__ agg,
                               float* __restrict__ cnt,
                               int do_cnt) {
    int e = __builtin_amdgcn_readfirstlane((int)(blockIdx.x * 8 + (threadIdx.x >> 5)));
    if (e >= NEDGES) return;                       // uniform
    const int lane = threadIdx.x & 31;
    const int src = ei[e];                         // scalar loads (uniform addr)
    const int dst = ei[NEDGES + e];
    const float4 v = *(const float4*)(x + (size_t)src * CH + lane * 4);
    float* a = agg + (size_t)dst * CH + lane * 4;
    unsafeAtomicAdd(a + 0, v.x);
    unsafeAtomicAdd(a + 1, v.y);
    unsafeAtomicAdd(a + 2, v.z);
    unsafeAtomicAdd(a + 3, v.w);
    if (do_cnt && lane == 0) unsafeAtomicAdd(&cnt[dst], 1.0f);
}

// ---------------- TDM: stage one 64KB weight matrix into LDS ----------------
// 1-row tile of 8192 x 8-byte elements, with hardware LDS padding:
// 16 dwords inserted every 128 dwords -> 576B effective row stride, which
// de-conflicts the 64-bank LDS for the WMMA B-fragment reads (2-row distance
// becomes 1152 % 256 = 128B -> opposite bank half for lanes 16-31).
// Descriptor per CDNA5 ISA ch.8:
//   group0: [1:0]count=1 | [63:32]lds_addr | [120:64]global_addr | [127:126]type=2
//   group1: [17:16]data_size=3(8B) | [20]pad_en | [24:22]pad_interval=6 (128 dw)
//           | [31:25]pad_amount=15 (16 dw) | tensor_dim0=tile_dim0=8192
__device__ __forceinline__ void tdm_stage(const float* gsrc, float* lds_dst) {
    unsigned long long ga = (unsigned long long)(uintptr_t)gsrc;
    unsigned int lds_off = (unsigned int)(uintptr_t)lds_dst;   // flat LDS addr low 32 = LDS offset
    uint32x4 g0;
    g0[0] = 1u;                                                 // count=1 (valid user D#)
    g0[1] = lds_off;                                            // lds_addr
    g0[2] = (unsigned int)ga;                                   // global_addr[31:0]
    g0[3] = (unsigned int)((ga >> 32) & 0x01FFFFFFu) | (2u << 30); // ga[56:32] | type=2
    int32x8 g1;
    g1[0] = (int)((3u << 16)           // data_size=3 (8 bytes), wg_mask=0
                | (1u << 20)           // pad_enable
                | (6u << 22)           // pad_interval: 2^(6+1)=128 dwords
                | (15u << 25));        // pad_amount: 15+1=16 dwords
    g1[1] = (int)(8192u << 16);        // tensor_dim0[15:0] = 8192
    g1[2] = (int)(1u << 16);           // tensor_dim0[31:16]=0, tensor_dim1=1
    g1[3] = (int)(8192u << 16);        // tile_dim0 = 8192
    g1[4] = 0;                         // tile_dim1/2 unused
    g1[5] = 8192;                      // tensor_dim0_stride (single row; benign)
    g1[6] = 0;
    g1[7] = 0;
    int32x4 z4 = {0, 0, 0, 0};         // groups 2/3 unused (<=2D)
    int32x8 z8 = {0, 0, 0, 0, 0, 0, 0, 0};
    __builtin_amdgcn_tensor_load_to_lds(g0, g1, z4, z4, z8, 0);
}

// ---------------- WMMA GEMM pass over one staged weight matrix --------------
// acc[nt] += (arow * scale?) @ sW, 16x16x4 f32 WMMA, K=CH, 8 column tiles.
// A-fragment for k+4 and B-fragment for nt+1 are prefetched one step ahead so
// the global/LDS waits overlap the matrix-pipe execution of the current WMMA.
template <bool SCALE>
__device__ __forceinline__ void gemm_pass(const float* __restrict__ arow, float scale,
                                          const float* __restrict__ sW,
                                          int half, int l16, v8f acc[8]) {
    const int koff = 2 * half;
    v2f an;
    an.x = arow[koff];
    an.y = arow[koff + 1];
    int b0 = koff * WSTRIDE + l16;
    v2f bn;
    bn.x = sW[b0];
    bn.y = sW[b0 + WSTRIDE];
    for (int k = 0; k < CH; k += 4) {
        v2f a = an;
        const int kn = (k + 4 < CH) ? (k + 4) : k;     // uniform clamp (no OOB read)
        an.x = arow[kn + koff];                         // prefetch next A fragment
        an.y = arow[kn + koff + 1];
        if (SCALE) { a.x *= scale; a.y *= scale; }      // -> v_pk_mul_f32
        const int cb = (k + koff) * WSTRIDE + l16;
        const int nb = (kn + koff) * WSTRIDE + l16;     // first frag of next k-step
#pragma unroll
        for (int nt = 0; nt < 8; ++nt) {
            v2f b = bn;
            const int pb = (nt < 7) ? (cb + (nt + 1) * 16) : nb;  // prefetch B
            bn.x = sW[pb];
            bn.y = sW[pb + WSTRIDE];
            acc[nt] = __builtin_amdgcn_wmma_f32_16x16x4_f32(
                false, a, false, b, (short)0, acc[nt], false, false);
        }
    }
}

// ---------------- fused dual-GEMM: out = act(A1*s @ W1 + A2 @ W2 + bias) ----
// Wave32 WMMA f32 16x16x4. One wave -> 16 output rows x 128 cols.
// Block = 8 waves = 128 rows. Weights TDM-staged in LDS (72KB padded layout).
__global__ __launch_bounds__(256) void fused_gemm_kernel(
    const float* __restrict__ A1, const float* __restrict__ cnt,
    const float* __restrict__ A2,
    const float* __restrict__ W1, const float* __restrict__ W2,
    const float* __restrict__ bias, float* __restrict__ out,
    int use_cnt, int do_relu)
{
    __shared__ float sW[CH * WSTRIDE];   // 72 KB of the 320 KB WGP LDS

    const int tid  = threadIdx.x;
    const int wave = tid >> 5;
    const int lane = tid & 31;
    const int half = lane >> 4;     // 0: lanes 0-15, 1: lanes 16-31
    const int l16  = lane & 15;

    const int row_tile     = (blockIdx.x * 8 + wave) * 16;
    const bool wave_active = row_tile < NNODES;   // uniform per wave

    // accumulators for 8 column tiles, bias folded in
    v8f acc[8];
#pragma unroll
    for (int nt = 0; nt < 8; ++nt) {
        float bv = bias[nt * 16 + l16];
#pragma unroll
        for (int j = 0; j < 8; ++j) acc[nt][j] = bv;
    }

    // A-fragment row for this lane (ISA: A 16x4 f32 -> M = lane&15)
    int rowM = row_tile + l16;
    if (rowM > NNODES - 1) rowM = NNODES - 1;     // clamp; pad rows never stored
    float scale1 = 1.0f;
    if (use_cnt) scale1 = 1.0f / fmaxf(cnt[rowM], 1.0f);

    // ---------- pass 1: TDM-stage W1, accumulate A1*scale1 @ W1 ----------
    if (wave == 0) {
        tdm_stage(W1, sW);
        __builtin_amdgcn_s_wait_tensorcnt(0);
    }
    __syncthreads();
    if (wave_active)
        gemm_pass<true>(A1 + (size_t)rowM * CH, scale1, sW, half, l16, acc);
    __syncthreads();

    // ---------- pass 2: TDM-stage W2, accumulate A2 @ W2, store ----------
    if (wave == 0) {
        tdm_stage(W2, sW);
        __builtin_amdgcn_s_wait_tensorcnt(0);
    }
    __syncthreads();
    if (wave_active) {
        gemm_pass<false>(A2 + (size_t)rowM * CH, 1.0f, sW, half, l16, acc);

        // D layout: VGPR j -> M = j + 8*half, N = lane&15
#pragma unroll
        for (int nt = 0; nt < 8; ++nt) {
#pragma unroll
            for (int j = 0; j < 8; ++j) {
                int r = row_tile + half * 8 + j;
                if (r < NNODES) {
                    float v = acc[nt][j];
                    if (do_relu) v = fmaxf(v, 0.0f);
                    out[(size_t)r * CH + nt * 16 + l16] = v;
                }
            }
        }
    }
}

extern "C" void kernel_launch(void* const* d_in, const int* in_sizes, int n_in,
                              void* d_out, int out_size, void* d_ws, size_t ws_size,
                              hipStream_t stream) {
    (void)in_sizes; (void)n_in; (void)out_size; (void)ws_size;

    const float* x     = (const float*)d_in[0];
    const int*   ei    = (const int*)  d_in[1];   // [2, E]
    const float* W1_l  = (const float*)d_in[2];
    const float* b1_l  = (const float*)d_in[3];
    const float* W1_r  = (const float*)d_in[4];
    const float* W2_l  = (const float*)d_in[5];
    const float* b2_l  = (const float*)d_in[6];
    const float* W2_r  = (const float*)d_in[7];
    const float* W_lin = (const float*)d_in[8];   // [256,128] row-major
    const float* b_lin = (const float*)d_in[9];
    float* out = (float*)d_out;

    // workspace layout: agg[N*CH] | cnt[N] | x1[N*CH] | x2[N*CH]
    float* agg = (float*)d_ws;
    float* cnt = agg + (size_t)NNODES * CH;
    float* x1  = cnt + NNODES;
    float* x2  = x1 + (size_t)NNODES * CH;

    const int TB = 256;
    const int n_aggcnt  = NNODES * CH + NNODES;           // agg+cnt contiguous
    const int n_agg     = NNODES * CH;
    const int scat_blks = (NEDGES + 7) / 8;               // 8 edges (waves) / block
    const int gemm_blks = ((NNODES + 15) / 16 + 7) / 8;   // 391

    // ---- layer 1 (degree folded into scatter) ----
    zero_f32<<<(n_aggcnt + TB - 1) / TB, TB, 0, stream>>>(agg, n_aggcnt);
    scatter_kernel<<<scat_blks, TB, 0, stream>>>(x, ei, agg, cnt, /*do_cnt=*/1);
    fused_gemm_kernel<<<gemm_blks, TB, 0, stream>>>(
        agg, cnt, x, W1_l, W1_r, b1_l, x1, /*use_cnt=*/1, /*relu=*/1);

    // ---- layer 2 (cnt reused) ----
    zero_f32<<<(n_agg + TB - 1) / TB, TB, 0, stream>>>(agg, n_agg);
    scatter_kernel<<<scat_blks, TB, 0, stream>>>(x1, ei, agg, cnt, /*do_cnt=*/0);
    fused_gemm_kernel<<<gemm_blks, TB, 0, stream>>>(
        agg, cnt, x1, W2_l, W2_r, b2_l, x2, /*use_cnt=*/1, /*relu=*/1);

    // ---- final linear: concat(x1,x2) @ W_lin + b_lin ----
    fused_gemm_kernel<<<gemm_blks, TB, 0, stream>>>(
        x1, nullptr, x2, W_lin, W_lin + CH * CH, b_lin, out,
        /*use_cnt=*/0, /*relu=*/0);
}